// MultiTurnPriorDecoder_Z_84482006712454
// MI455X (gfx1250) — compile-verified
//
#include <hip/hip_runtime.h>
#include <hip/hip_bf16.h>
#include <math.h>

// Problem constants (match reference)
#define B_  64
#define TU_ 128
#define TZ_ 16
#define V_  32000
#define H_  512
#define E_  256
#define EPS_ 1e-10f
#define SPLITK_ 32          // K-chunks for the K=32000 GEMMs (chunk = 1000)

typedef __attribute__((ext_vector_type(2))) float v2f;
typedef __attribute__((ext_vector_type(8))) float v8f;

#define WMMA_F32(a, b, c) \
    __builtin_amdgcn_wmma_f32_16x16x4_f32(false, (a), false, (b), (short)0, (c), false, false)

// ---------------------------------------------------------------------------
// Generic WMMA fp32 GEMM:  C[m,n] = act(bias[n] + sum_k A[m,k] * W[n,k])
// One wave (32 lanes) computes a 16x64 C tile (4 independent WMMA chains)
// via V_WMMA_F32_16X16X4_F32.
// grid.x = N/64 tiles, grid.y = M/16 tiles, grid.z = batch index OR k-chunk.
// kChunk==0 : full-K, z is a batch index (strideAz/strideCz are batch strides)
// kChunk>0  : split-K, z selects K range [z*kChunk, z*kChunk+kChunk); partials
//             are written to C + z*strideCz (bias must be null, act must be 0)
// ---------------------------------------------------------------------------
__global__ void __launch_bounds__(32)
wmma_gemm_awt(const float* __restrict__ A, const float* __restrict__ W,
              const float* __restrict__ bias, float* __restrict__ C,
              int M, int N, int K,
              long lda, long ldw, long ldc,
              long strideAz, long strideCz, int kChunk, int act_tanh)
{
    const int lane = threadIdx.x;
    const int half = lane >> 4;          // 0: lanes 0-15, 1: lanes 16-31
    const int lh   = lane & 15;
    const int n0   = blockIdx.x * 64;
    const int m0   = blockIdx.y * 16;
    const int z    = blockIdx.z;

    const float* Ab = A;
    float*       Cb = C;
    int k0 = 0, k1 = K;
    if (kChunk > 0) {
        k0 = z * kChunk;
        k1 = k0 + kChunk; if (k1 > K) k1 = K;
        Cb = C + (long)z * strideCz;
    } else {
        Ab = A + (long)z * strideAz;
        Cb = C + (long)z * strideCz;
    }

    // A 16x4 layout: lanes 0-15 hold (M=lh, K=k..k+1); lanes 16-31 hold K=k+2..k+3
    // B 4x16 layout mirrors it with N=lh.  4 N-subtiles -> 4 B-row streams.
    const float* ap  = Ab + (long)(m0 + lh) * lda + k0 + 2 * half;
    const float* wp0 = W  + (long)(n0      + lh) * ldw + k0 + 2 * half;
    const float* wp1 = W  + (long)(n0 + 16 + lh) * ldw + k0 + 2 * half;
    const float* wp2 = W  + (long)(n0 + 32 + lh) * ldw + k0 + 2 * half;
    const float* wp3 = W  + (long)(n0 + 48 + lh) * ldw + k0 + 2 * half;

    v8f acc[4];
    acc[0] = (v8f){}; acc[1] = (v8f){}; acc[2] = (v8f){}; acc[3] = (v8f){};

    int k = k0;
    for (; k + 8 <= k1; k += 8) {
        // stream-ahead hints on the compulsory W traffic (global_prefetch_b8)
        __builtin_prefetch(wp0 + 1024, 0, 3);
        __builtin_prefetch(wp1 + 1024, 0, 3);
        __builtin_prefetch(wp2 + 1024, 0, 3);
        __builtin_prefetch(wp3 + 1024, 0, 3);

        v2f aA = *(const v2f*)(ap);
        v2f b0 = *(const v2f*)(wp0);
        v2f b1 = *(const v2f*)(wp1);
        v2f b2 = *(const v2f*)(wp2);
        v2f b3 = *(const v2f*)(wp3);
        acc[0] = WMMA_F32(aA, b0, acc[0]);
        acc[1] = WMMA_F32(aA, b1, acc[1]);
        acc[2] = WMMA_F32(aA, b2, acc[2]);
        acc[3] = WMMA_F32(aA, b3, acc[3]);

        v2f aB = *(const v2f*)(ap  + 4);
        v2f c0 = *(const v2f*)(wp0 + 4);
        v2f c1 = *(const v2f*)(wp1 + 4);
        v2f c2 = *(const v2f*)(wp2 + 4);
        v2f c3 = *(const v2f*)(wp3 + 4);
        acc[0] = WMMA_F32(aB, c0, acc[0]);
        acc[1] = WMMA_F32(aB, c1, acc[1]);
        acc[2] = WMMA_F32(aB, c2, acc[2]);
        acc[3] = WMMA_F32(aB, c3, acc[3]);

        ap += 8; wp0 += 8; wp1 += 8; wp2 += 8; wp3 += 8;
    }
    for (; k < k1; k += 4) {
        v2f aA = *(const v2f*)(ap);
        v2f b0 = *(const v2f*)(wp0);
        v2f b1 = *(const v2f*)(wp1);
        v2f b2 = *(const v2f*)(wp2);
        v2f b3 = *(const v2f*)(wp3);
        acc[0] = WMMA_F32(aA, b0, acc[0]);
        acc[1] = WMMA_F32(aA, b1, acc[1]);
        acc[2] = WMMA_F32(aA, b2, acc[2]);
        acc[3] = WMMA_F32(aA, b3, acc[3]);
        ap += 4; wp0 += 4; wp1 += 4; wp2 += 4; wp3 += 4;
    }

#pragma unroll
    for (int j = 0; j < 4; ++j) {
        const int n = n0 + 16 * j + lh;
        const float bv = (bias != nullptr && kChunk == 0) ? bias[n] : 0.0f;
#pragma unroll
        for (int r = 0; r < 8; ++r) {
            // C/D layout: VGPR r -> M = r (lanes 0-15) or r+8 (lanes 16-31), N = lh
            const int m = m0 + r + 8 * half;
            float v = acc[j][r] + bv;
            if (act_tanh) v = tanhf(v);
            Cb[(long)m * ldc + n] = v;
        }
    }
}

// ---------------------------------------------------------------------------
// Single GRU step: h = (1-z)*n + z*h0  (one thread per (b,j) output element)
// ---------------------------------------------------------------------------
__global__ void __launch_bounds__(256)
gru_kernel(const float* __restrict__ x,     // (B,E)
           const float* __restrict__ h0,    // (B,H)
           const float* __restrict__ W_ih,  // (3H,E)
           const float* __restrict__ W_hh,  // (3H,H)
           const float* __restrict__ b_ih, const float* __restrict__ b_hh,
           float* __restrict__ h, float* __restrict__ out_h1, float* __restrict__ out_h2)
{
    const int idx = blockIdx.x * blockDim.x + threadIdx.x;  // < B*H
    const int b = idx / H_;
    const int j = idx % H_;

    float gr = b_ih[j], gz = b_ih[H_ + j], gn = b_ih[2 * H_ + j];
    const float* xb = x + (long)b * E_;
    const float* wr = W_ih + (long)j * E_;
    const float* wz = W_ih + (long)(H_ + j) * E_;
    const float* wn = W_ih + (long)(2 * H_ + j) * E_;
    for (int e = 0; e < E_; ++e) {
        const float xv = xb[e];
        gr += xv * wr[e]; gz += xv * wz[e]; gn += xv * wn[e];
    }
    float hr = b_hh[j], hz = b_hh[H_ + j], hn = b_hh[2 * H_ + j];
    const float* hb = h0 + (long)b * H_;
    const float* vr = W_hh + (long)j * H_;
    const float* vz = W_hh + (long)(H_ + j) * H_;
    const float* vn = W_hh + (long)(2 * H_ + j) * H_;
    for (int k = 0; k < H_; ++k) {
        const float hv = hb[k];
        hr += hv * vr[k]; hz += hv * vz[k]; hn += hv * vn[k];
    }
    const float r  = 1.0f / (1.0f + expf(-(gr + hr)));
    const float zg = 1.0f / (1.0f + expf(-(gz + hz)));
    const float nn = tanhf(gn + r * hn);
    const float hv = (1.0f - zg) * nn + zg * hb[j];
    h[idx] = hv; out_h1[idx] = hv; out_h2[idx] = hv;
}

// ---------------------------------------------------------------------------
// out[row] = (exp?) dot(Amat[row,:], hvec[row/T,:])   one wave per row
// ---------------------------------------------------------------------------
__global__ void __launch_bounds__(32)
rowdot_kernel(const float* __restrict__ Amat, const float* __restrict__ hvec,
              float* __restrict__ out, int T, int H, int apply_exp)
{
    const int row = blockIdx.x;           // b*T + t
    const int b = row / T;
    const float* a  = Amat + (long)row * H;
    const float* hv = hvec + (long)b * H;
    float acc = 0.0f;
    for (int k = threadIdx.x; k < H; k += 32) acc += a[k] * hv[k];
    for (int off = 16; off > 0; off >>= 1) acc += __shfl_down(acc, off, 32);
    if (threadIdx.x == 0) out[row] = apply_exp ? expf(acc) : acc;
}

// ---------------------------------------------------------------------------
// Per-row stable-exp stats for u_copy: um=max, ue=exp(us-um), usum=sum(ue)
// ---------------------------------------------------------------------------
__global__ void __launch_bounds__(128)
ucopy_stats_kernel(const float* __restrict__ us, float* __restrict__ ue,
                   float* __restrict__ um, float* __restrict__ usum)
{
    __shared__ float sm[128];
    const int b = blockIdx.x, t = threadIdx.x;
    const float v = us[b * TU_ + t];
    sm[t] = v; __syncthreads();
    for (int s = 64; s > 0; s >>= 1) { if (t < s) sm[t] = fmaxf(sm[t], sm[t + s]); __syncthreads(); }
    const float m = sm[0]; __syncthreads();
    const float e = expf(v - m);
    ue[b * TU_ + t] = e;
    sm[t] = e; __syncthreads();
    for (int s = 64; s > 0; s >>= 1) { if (t < s) sm[t] += sm[t + s]; __syncthreads(); }
    if (t == 0) { um[b] = m; usum[b] = sm[0]; }
}

__global__ void __launch_bounds__(256)
fill_ulin_kernel(float* __restrict__ u_lin, const float* __restrict__ usum)
{
    const long i = (long)blockIdx.x * blockDim.x + threadIdx.x;  // < B*V
    const int b = (int)(i / V_);
    u_lin[i] = EPS_ * usum[b];
}

// Deterministic scatter-add (one thread per batch row, serial over tokens)
__global__ void __launch_bounds__(64)
scatter_kernel(const int* __restrict__ tok /*(TU,B)*/,
               const float* __restrict__ ue, float* __restrict__ u_lin)
{
    const int b = threadIdx.x;  // 64 threads
    float* row = u_lin + (long)b * V_;
    for (int t = 0; t < TU_; ++t) {
        const int tk = tok[t * B_ + b];
        if (tk != 0) row[tk] += ue[b * TU_ + t];
    }
}

// pv_copy[b,v] = log( ps[b,0]*EPS + sum_{t=1..TZ-1} ps[b,t]*pz[t-1,b,v] )
__global__ void __launch_bounds__(256)
pvcopy_kernel(const float* __restrict__ pz /*(TZ,B,V)*/,
              const float* __restrict__ ps /*(B,TZ)*/,
              float* __restrict__ pv /*(B,V)*/)
{
    const int v = blockIdx.x * blockDim.x + threadIdx.x;  // < V
    const int b = blockIdx.y;
    float acc = ps[b * TZ_ + 0] * EPS_;
#pragma unroll
    for (int t = 1; t < TZ_; ++t)
        acc += ps[b * TZ_ + t] * pz[((long)(t - 1) * B_ + b) * V_ + v];
    pv[(long)b * V_ + v] = logf(acc);
}

// Joint softmax over [gen | u_copy | pv_copy] per row, then merge to proba
__global__ void __launch_bounds__(256)
merge_softmax_kernel(const float* __restrict__ gen, const float* __restrict__ u_lin,
                     const float* __restrict__ um, const float* __restrict__ pv,
                     float* __restrict__ proba)
{
    __shared__ float red[256];
    const int b = blockIdx.x, t = threadIdx.x;
    const long off = (long)b * V_;
    const float mu = um[b];

    float m = -3.4e38f;
    for (int v = t; v < V_; v += 256) {
        m = fmaxf(m, gen[off + v]);
        m = fmaxf(m, logf(u_lin[off + v]) + mu);
        m = fmaxf(m, pv[off + v]);
    }
    red[t] = m; __syncthreads();
    for (int s = 128; s > 0; s >>= 1) { if (t < s) red[t] = fmaxf(red[t], red[t + s]); __syncthreads(); }
    m = red[0]; __syncthreads();

    const float su = expf(mu - m);   // exp(log(u_lin)+mu-m) == u_lin*su
    float sum = 0.0f;
    for (int v = t; v < V_; v += 256)
        sum += expf(gen[off + v] - m) + u_lin[off + v] * su + expf(pv[off + v] - m);
    red[t] = sum; __syncthreads();
    for (int s = 128; s > 0; s >>= 1) { if (t < s) red[t] += red[t + s]; __syncthreads(); }
    const float inv = 1.0f / red[0];

    for (int v = t; v < V_; v += 256) {
        const float p = expf(gen[off + v] - m) + u_lin[off + v] * su + expf(pv[off + v] - m);
        proba[off + v] = p * inv;
    }
}

// Reduce split-K partials + bias, and VAE reparameterization
__global__ void __launch_bounds__(256)
finalize_kernel(const float* __restrict__ pa, const float* __restrict__ pb,
                const float* __restrict__ b_mu, const float* __restrict__ b_ls,
                const float* __restrict__ eps,
                float* __restrict__ appr, float* __restrict__ ls_out,
                float* __restrict__ sampled)
{
    const int i = blockIdx.x * blockDim.x + threadIdx.x;  // < B*E
    const int e = i % E_;
    float a = b_mu[e], l = b_ls[e];
#pragma unroll
    for (int s = 0; s < SPLITK_; ++s) { a += pa[s * (B_ * E_) + i]; l += pb[s * (B_ * E_) + i]; }
    appr[i] = a; ls_out[i] = l;
    sampled[i] = a + expf(l) * eps[i];
}

// ---------------------------------------------------------------------------
extern "C" void kernel_launch(void* const* d_in, const int* in_sizes, int n_in,
                              void* d_out, int out_size, void* d_ws, size_t ws_size,
                              hipStream_t stream)
{
    const float* embed_z     = (const float*)d_in[0];
    const float* last_hidden = (const float*)d_in[1];
    const float* u_enc_out   = (const float*)d_in[2];
    const float* pv_pz       = (const float*)d_in[3];
    const float* pv_z_dec    = (const float*)d_in[4];
    const float* rand_eps    = (const float*)d_in[5];
    const int*   u_input_np  = (const int*)d_in[6];
    // d_in[7], d_in[8] unused by forward
    const float* W_ih = (const float*)d_in[9];
    const float* W_hh = (const float*)d_in[10];
    const float* b_ih = (const float*)d_in[11];
    const float* b_hh = (const float*)d_in[12];
    const float* W_w1 = (const float*)d_in[13];
    const float* b_w1 = (const float*)d_in[14];
    const float* W_p1 = (const float*)d_in[15];
    const float* b_p1 = (const float*)d_in[16];
    const float* W_p2 = (const float*)d_in[17];
    const float* b_p2 = (const float*)d_in[18];
    const float* W_mu = (const float*)d_in[19];
    const float* b_mu = (const float*)d_in[20];
    const float* W_ls = (const float*)d_in[21];
    const float* b_ls = (const float*)d_in[22];

    // Output layout (flat, tuple order): sampled(16384), h(32768), h(32768),
    // proba(2048000), appr(16384), log_sigma(16384)
    float* out       = (float*)d_out;
    float* o_sampled = out;
    float* o_h1      = out + 16384;
    float* o_h2      = out + 49152;
    float* o_proba   = out + 81920;
    float* o_appr    = out + 2129920;
    float* o_ls      = out + 2146304;

    // Workspace layout (floats)
    float* ws     = (float*)d_ws;
    float* w_h    = ws;                    // B*H          = 32768
    float* w_gen  = w_h    + 32768;        // B*V          = 2048000
    float* w_uc   = w_gen  + 2048000;      // B*TU*H       = 4194304
    float* w_us   = w_uc   + 4194304;      // B*TU         = 8192
    float* w_ue   = w_us   + 8192;         // B*TU         = 8192
    float* w_um   = w_ue   + 8192;         // B            = 64
    float* w_usum = w_um   + 64;           // B            = 64
    float* w_ulin = w_usum + 64;           // B*V          = 2048000
    float* w_pc   = w_ulin + 2048000;      // B*TZ*H       = 524288
    float* w_ps   = w_pc   + 524288;       // B*TZ         = 1024
    float* w_pv   = w_ps   + 1024;         // B*V          = 2048000
    float* w_pa   = w_pv   + 2048000;      // SPLITK*B*E   = 524288
    float* w_pb   = w_pa   + 524288;       // SPLITK*B*E   = 524288

    // 1) GRU step -> h (also written to the two h output slots)
    gru_kernel<<<128, 256, 0, stream>>>(embed_z, last_hidden, W_ih, W_hh, b_ih, b_hh,
                                        w_h, o_h1, o_h2);

    // 2) gen_score = h @ W_w1^T + b_w1            (M=64, N=32000, K=512)
    wmma_gemm_awt<<<dim3(V_ / 64, B_ / 16, 1), 32, 0, stream>>>(
        w_h, W_w1, b_w1, w_gen, B_, V_, H_, H_, H_, V_, 0, 0, 0, 0);

    // 3) uc = tanh(u_enc^T @ W_p1^T + b_p1)       batched over b (M=128,N=512,K=512)
    wmma_gemm_awt<<<dim3(H_ / 64, TU_ / 16, B_), 32, 0, stream>>>(
        u_enc_out, W_p1, b_p1, w_uc, TU_, H_, H_,
        (long)B_ * H_, H_, H_, H_, (long)TU_ * H_, 0, 1);

    // 4) us[b,t] = dot(uc[b,t,:], h[b,:])
    rowdot_kernel<<<B_ * TU_, 32, 0, stream>>>(w_uc, w_h, w_us, TU_, H_, 0);

    // 5) stable exp stats; linear u_copy fill; deterministic token scatter
    ucopy_stats_kernel<<<B_, 128, 0, stream>>>(w_us, w_ue, w_um, w_usum);
    fill_ulin_kernel<<<(B_ * V_) / 256, 256, 0, stream>>>(w_ulin, w_usum);
    scatter_kernel<<<1, 64, 0, stream>>>(u_input_np, w_ue, w_ulin);

    // 6) pc = tanh(pv_z_dec^T @ W_p2^T + b_p2)    batched over b (M=16,N=512,K=512)
    wmma_gemm_awt<<<dim3(H_ / 64, TZ_ / 16, B_), 32, 0, stream>>>(
        pv_z_dec, W_p2, b_p2, w_pc, TZ_, H_, H_,
        (long)B_ * H_, H_, H_, H_, (long)TZ_ * H_, 0, 1);

    // 7) ps[b,t] = exp(dot(pc[b,t,:], h[b,:]))
    rowdot_kernel<<<B_ * TZ_, 32, 0, stream>>>(w_pc, w_h, w_ps, TZ_, H_, 1);

    // 8) pv_copy (131 MB coalesced stream of pv_pz_proba)
    pvcopy_kernel<<<dim3(V_ / 256, B_), 256, 0, stream>>>(pv_pz, w_ps, w_pv);

    // 9) joint softmax over [gen | u_copy | pv_copy] -> proba
    merge_softmax_kernel<<<B_, 256, 0, stream>>>(w_gen, w_ulin, w_um, w_pv, o_proba);

    // 10) appr / log_sigma = proba @ W^T  (split-K=32 for parallelism, deterministic)
    wmma_gemm_awt<<<dim3(E_ / 64, B_ / 16, SPLITK_), 32, 0, stream>>>(
        o_proba, W_mu, nullptr, w_pa, B_, E_, V_, V_, V_, E_, 0, (long)B_ * E_, V_ / SPLITK_, 0);
    wmma_gemm_awt<<<dim3(E_ / 64, B_ / 16, SPLITK_), 32, 0, stream>>>(
        o_proba, W_ls, nullptr, w_pb, B_, E_, V_, V_, V_, E_, 0, (long)B_ * E_, V_ / SPLITK_, 0);

    // 11) reduce partials + bias, VAE reparameterization
    finalize_kernel<<<(B_ * E_) / 256, 256, 0, stream>>>(
        w_pa, w_pb, b_mu, b_ls, rand_eps, o_appr, o_ls, o_sampled);
}